// MultiLayerGCN_90280212562156
// MI455X (gfx1250) — compile-verified
//
#include <hip/hip_runtime.h>

#define N_NODES 50000
#define N_EDGES 800000
#define F_IN    128
#define F_EMB1  512
#define F_EMB2  128
#define F_HID   1024
#define F_OUT   64

typedef float v2f __attribute__((ext_vector_type(2)));
typedef float v8f __attribute__((ext_vector_type(8)));

// ---------------- graph-normalization kernels ----------------

__global__ void deg_scatter(const int* __restrict__ dst, const float* __restrict__ w,
                            float* __restrict__ deg, int E) {
    int e = blockIdx.x * blockDim.x + threadIdx.x;
    if (e < E) atomicAdd(&deg[dst[e]], w[e]);
}

__global__ void node_norm(const float* __restrict__ deg, float* __restrict__ dinv,
                          float* __restrict__ selfinv, int n) {
    int i = blockIdx.x * blockDim.x + threadIdx.x;
    if (i < n) {
        float d = deg[i] + 1.0f;           // +1 self-loop weight
        dinv[i]    = rsqrtf(d);
        selfinv[i] = 1.0f / d;
    }
}

__global__ void edge_norm(const int* __restrict__ src, const int* __restrict__ dst,
                          const float* __restrict__ w, const float* __restrict__ dinv,
                          float* __restrict__ norm, int E) {
    int e = blockIdx.x * blockDim.x + threadIdx.x;
    if (e < E) norm[e] = dinv[src[e]] * w[e] * dinv[dst[e]];
}

// ---------------- edge aggregation: agg[dst] += norm * h[src] ----------------
// one block per edge; lanes cover features in float4; src/dst/norm are
// block-uniform -> scalar loads.

__global__ void edge_scatter(const int* __restrict__ src, const int* __restrict__ dst,
                             const float* __restrict__ norm, const float* __restrict__ h,
                             float* __restrict__ agg, int F) {
    int e = blockIdx.x;
    int s = src[e];
    int d = dst[e];
    float nw = norm[e];
    int f = threadIdx.x << 2;
    float4 hv = *(const float4*)(h + (long)s * F + f);
    float* out = agg + (long)d * F + f;
    atomicAdd(out + 0, nw * hv.x);
    atomicAdd(out + 1, nw * hv.y);
    atomicAdd(out + 2, nw * hv.z);
    atomicAdd(out + 3, nw * hv.w);
}

// out = relu(agg + h*selfinv[row] + bias), in place on agg
__global__ void combine_bias_relu(float* __restrict__ agg, const float* __restrict__ h,
                                  const float* __restrict__ selfinv,
                                  const float* __restrict__ bias, int F) {
    int row = blockIdx.x;
    float si = selfinv[row];
    int f = threadIdx.x << 2;
    float4 a  = *(float4*)(agg + (long)row * F + f);
    float4 hv = *(const float4*)(h + (long)row * F + f);
    float4 b  = *(const float4*)(bias + f);
    a.x = fmaxf(fmaf(hv.x, si, a.x) + b.x, 0.0f);
    a.y = fmaxf(fmaf(hv.y, si, a.y) + b.y, 0.0f);
    a.z = fmaxf(fmaf(hv.z, si, a.z) + b.z, 0.0f);
    a.w = fmaxf(fmaf(hv.w, si, a.w) + b.w, 0.0f);
    *(float4*)(agg + (long)row * F + f) = a;
}

// ---------------- FP32 WMMA GEMM: C = A[M,K] @ B[K,N] (+bias, relu) ----------
// blockDim = (32, 8): 8 waves. Each wave owns a 2x4 grid of 16x16 tiles
// (32 rows x 64 cols): B fragments are loaded once and reused by both A
// fragments -> 8 v_wmma_f32_16x16x4_f32 per 10 global loads in steady state.
// A frag (16x4 f32, 2 VGPRs): lane 0-15 -> K=0,1 ; lane 16-31 -> K=2,3.
// B frag (4x16): v0 = row K+kh, v1 = row K+kh+1, col = lane&15.
// C/D (8 VGPRs): col = lane&15, row = vgpr + 8*(lane>>4).

#define WMMA_F32(A_, B_, C_) \
    __builtin_amdgcn_wmma_f32_16x16x4_f32(false, (A_), false, (B_), (short)0, (C_), false, false)

template <int BIAS, int RELU>
__global__ __launch_bounds__(256)
void gemm_wmma(const float* __restrict__ A, const float* __restrict__ B,
               const float* __restrict__ bias, float* __restrict__ C,
               int M, int K, int Nout) {
    const int lane   = threadIdx.x;                  // 0..31
    const int wave   = threadIdx.y;                  // 0..7
    const int pair   = blockIdx.y * 8 + wave;        // pair of M-tiles
    const int tileM0 = pair * 2;
    if (tileM0 * 16 >= M) return;                    // wave-uniform exit
    const int  tileM1 = tileM0 + 1;
    const bool has1   = (tileM1 * 16) < M;
    const int  n_base = blockIdx.x * 64;             // 4 N-tiles of 16
    const int  lm = lane & 15;
    const int  kh = (lane >> 4) << 1;                // 0 or 2
    const long arow0 = (long)(tileM0 * 16 + lm) * K;
    const long arow1 = has1 ? (long)(tileM1 * 16 + lm) * K : arow0;  // clamp: computed, never stored

    v8f acc00 = {}, acc01 = {}, acc02 = {}, acc03 = {};
    v8f acc10 = {}, acc11 = {}, acc12 = {}, acc13 = {};
    for (int k = 0; k < K; k += 4) {
        v2f a0 = *(const v2f*)(A + arow0 + k + kh);
        v2f a1 = *(const v2f*)(A + arow1 + k + kh);
        const float* b0p = B + (long)(k + kh) * Nout + n_base + lm;
        const float* b1p = b0p + Nout;
        v2f b0, b1, b2, b3;
        b0.x = b0p[0];  b0.y = b1p[0];
        b1.x = b0p[16]; b1.y = b1p[16];
        b2.x = b0p[32]; b2.y = b1p[32];
        b3.x = b0p[48]; b3.y = b1p[48];
        acc00 = WMMA_F32(a0, b0, acc00);
        acc10 = WMMA_F32(a1, b0, acc10);
        acc01 = WMMA_F32(a0, b1, acc01);
        acc11 = WMMA_F32(a1, b1, acc11);
        acc02 = WMMA_F32(a0, b2, acc02);
        acc12 = WMMA_F32(a1, b2, acc12);
        acc03 = WMMA_F32(a0, b3, acc03);
        acc13 = WMMA_F32(a1, b3, acc13);
    }

    const int rhalf = (lane >> 4) << 3;              // 0 or 8
#define STORE_TILE(ACC, TM, T)                                               \
    {                                                                        \
        int n = n_base + (T) * 16 + lm;                                      \
        int rbase = (TM) * 16 + rhalf;                                       \
        float bi = BIAS ? bias[n] : 0.0f;                                    \
        _Pragma("unroll") for (int i = 0; i < 8; ++i) {                      \
            float v = ACC[i] + bi;                                           \
            if (RELU) v = fmaxf(v, 0.0f);                                    \
            C[(long)(rbase + i) * Nout + n] = v;                             \
        }                                                                    \
    }
    STORE_TILE(acc00, tileM0, 0)
    STORE_TILE(acc01, tileM0, 1)
    STORE_TILE(acc02, tileM0, 2)
    STORE_TILE(acc03, tileM0, 3)
    if (has1) {
        STORE_TILE(acc10, tileM1, 0)
        STORE_TILE(acc11, tileM1, 1)
        STORE_TILE(acc12, tileM1, 2)
        STORE_TILE(acc13, tileM1, 3)
    }
#undef STORE_TILE
}

// ---------------- launch ----------------

extern "C" void kernel_launch(void* const* d_in, const int* in_sizes, int n_in,
                              void* d_out, int out_size, void* d_ws, size_t ws_size,
                              hipStream_t stream) {
    (void)in_sizes; (void)n_in; (void)out_size; (void)ws_size;
    const float* x  = (const float*)d_in[0];
    const int*   ei = (const int*)d_in[1];
    const float* ew = (const float*)d_in[2];
    const float* W1 = (const float*)d_in[3];
    const float* b1 = (const float*)d_in[4];
    const float* W2 = (const float*)d_in[5];
    const float* b2 = (const float*)d_in[6];
    const float* W3 = (const float*)d_in[7];
    const float* b3 = (const float*)d_in[8];
    const float* W4 = (const float*)d_in[9];
    const float* b4 = (const float*)d_in[10];
    const int* src = ei;
    const int* dst = ei + N_EDGES;

    float* ws = (float*)d_ws;
    size_t off = 0;
    auto alloc = [&](size_t nf) {
        float* p = ws + off;
        off += (nf + 63) & ~(size_t)63;
        return p;
    };
    float* deg     = alloc(N_NODES);
    float* dinv    = alloc(N_NODES);
    float* selfinv = alloc(N_NODES);
    float* norm    = alloc(N_EDGES);
    float* bufA    = alloc((size_t)N_NODES * F_HID);   // h1 [,512] then h3 [,1024]
    float* bufB    = alloc((size_t)N_NODES * F_EMB1);  // agg1 -> out1
    float* bufC    = alloc((size_t)N_NODES * F_EMB2);  // h2
    float* bufD    = alloc((size_t)N_NODES * F_EMB2);  // agg2 -> out2

    // ---- degree / normalization ----
    hipMemsetAsync(deg, 0, N_NODES * sizeof(float), stream);
    deg_scatter<<<(N_EDGES + 255) / 256, 256, 0, stream>>>(dst, ew, deg, N_EDGES);
    node_norm  <<<(N_NODES + 255) / 256, 256, 0, stream>>>(deg, dinv, selfinv, N_NODES);
    edge_norm  <<<(N_EDGES + 255) / 256, 256, 0, stream>>>(src, dst, ew, dinv, norm, N_EDGES);

    const dim3 blk(32, 8);
    const int MTILES = N_NODES / 16;                 // 3125 (exact)
    const int MPAIRS = (MTILES + 1) / 2;             // 1563
    const int GY = (MPAIRS + 7) / 8;                 // 196

    // ---- layer 1: h1 = x @ W1 ; agg ; relu(+b1) ----
    gemm_wmma<0, 0><<<dim3(F_EMB1 / 64, GY), blk, 0, stream>>>(x, W1, nullptr, bufA,
                                                               N_NODES, F_IN, F_EMB1);
    hipMemsetAsync(bufB, 0, (size_t)N_NODES * F_EMB1 * sizeof(float), stream);
    edge_scatter<<<N_EDGES, F_EMB1 / 4, 0, stream>>>(src, dst, norm, bufA, bufB, F_EMB1);
    combine_bias_relu<<<N_NODES, F_EMB1 / 4, 0, stream>>>(bufB, bufA, selfinv, b1, F_EMB1);

    // ---- layer 2: h2 = out1 @ W2 ; agg ; relu(+b2) ----
    gemm_wmma<0, 0><<<dim3(F_EMB2 / 64, GY), blk, 0, stream>>>(bufB, W2, nullptr, bufC,
                                                               N_NODES, F_EMB1, F_EMB2);
    hipMemsetAsync(bufD, 0, (size_t)N_NODES * F_EMB2 * sizeof(float), stream);
    edge_scatter<<<N_EDGES, F_EMB2 / 4, 0, stream>>>(src, dst, norm, bufC, bufD, F_EMB2);
    combine_bias_relu<<<N_NODES, F_EMB2 / 4, 0, stream>>>(bufD, bufC, selfinv, b2, F_EMB2);

    // ---- layer 3: h3 = relu(out2 @ W3 + b3) ----
    gemm_wmma<1, 1><<<dim3(F_HID / 64, GY), blk, 0, stream>>>(bufD, W3, b3, bufA,
                                                              N_NODES, F_EMB2, F_HID);

    // ---- layer 4: out = relu(h3 @ W4 + b4) ----
    gemm_wmma<1, 1><<<dim3(F_OUT / 64, GY), blk, 0, stream>>>(bufA, W4, b4, (float*)d_out,
                                                              N_NODES, F_HID, F_OUT);
}